// GCN_71811853189639
// MI455X (gfx1250) — compile-verified
//
#include <hip/hip_runtime.h>
#include <hip/hip_bf16.h>
#include <math.h>

// ---------------------------------------------------------------------------
// GCN forward for MI455X (gfx1250, wave32, WMMA).
// Precision: f32 WMMA 16x16x4 (reference is fp32 and feeds an argmax; bf16
// input rounding could flip node selection). Dominant GEMM2 is 34.4 GFLOP;
// memory floor ~29us @ 23.3TB/s, f32 matrix path keeps us within a small
// factor of that while matching reference numerics.
// ---------------------------------------------------------------------------

typedef __attribute__((ext_vector_type(2))) float v2f;
typedef __attribute__((ext_vector_type(8))) float v8f;

#define WMMA_F32X4(A, B, C) \
    __builtin_amdgcn_wmma_f32_16x16x4_f32(false, (A), false, (B), (short)0, (C), false, false)

static constexpr int Bn = 256;   // batch
static constexpr int Nn = 512;   // nodes
static constexpr int Fn = 256;   // in features
static constexpr int Hn = 256;   // hidden
static constexpr int NSTAGE = 5;
static constexpr float EPS = 1e-5f;

// ---------------------------------------------------------------------------
// Kernel 1: support[b,n,h] = features[b,n,:] @ W0[:,h]
// grid = (B*N)/16 blocks of 256 threads (8 waves). Each wave owns two 16-col
// tiles; K loop of 64 steps of V_WMMA_F32_16X16X4_F32.
// ---------------------------------------------------------------------------
__global__ __launch_bounds__(256) void gemm1_kernel(
    const float* __restrict__ features,   // [B*N, F]
    const float* __restrict__ W0,         // [F, H]
    float* __restrict__ support)          // [B*N, H]
{
    constexpr int APAD = 260;             // 256 + 4 pad -> conflict-free column reads
    __shared__ float As[16 * APAD];

    const int tid  = threadIdx.x;
    const long m0  = (long)blockIdx.x * 16;

    // stage 16x256 A tile into LDS (rows contiguous in memory)
    const float4* src = (const float4*)(features + m0 * Fn);
    for (int i = tid; i < 16 * (Fn / 4); i += 256) {
        int r = i >> 6, c4 = i & 63;
        ((float4*)(As + r * APAD))[c4] = src[i];
    }
    __syncthreads();

    const int wave  = tid >> 5;
    const int lane  = tid & 31;
    const int row   = lane & 15;
    const int khalf = lane >> 4;          // 0: K={0,1}  1: K={2,3}
    const int colA  = wave * 32 + (lane & 15);
    const int colB  = colA + 16;

    v8f acc0 = {}; v8f acc1 = {};
    #pragma unroll 4
    for (int k0 = 0; k0 < Fn; k0 += 4) {
        const int ka = k0 + 2 * khalf;
        v2f a;  a.x  = As[row * APAD + ka];
                a.y  = As[row * APAD + ka + 1];
        v2f b0v; b0v.x = W0[ka * Hn + colA];
                 b0v.y = W0[(ka + 1) * Hn + colA];
        v2f b1v; b1v.x = W0[ka * Hn + colB];
                 b1v.y = W0[(ka + 1) * Hn + colB];
        acc0 = WMMA_F32X4(a, b0v, acc0);
        acc1 = WMMA_F32X4(a, b1v, acc1);
    }

    // C/D layout: VGPR r -> row r (lanes 0-15) / row r+8 (lanes 16-31)
    const int crow = (lane < 16) ? 0 : 8;
    const int ccol = lane & 15;
    #pragma unroll
    for (int r = 0; r < 8; ++r) {
        const long rr = m0 + crow + r;
        support[rr * Hn + wave * 32 + ccol]      = acc0[r];
        support[rr * Hn + wave * 32 + 16 + ccol] = acc1[r];
    }
}

// ---------------------------------------------------------------------------
// Kernel 2: x = adjacency[b] @ support[b] + b0 ; LN(g0,be0) ; ReLU ; +b1 ;
//           scores[b,n] = row sum. Only scores are written.
// grid = B * (N/16) blocks of 256 threads.
// ---------------------------------------------------------------------------
__global__ __launch_bounds__(256) void gemm2_scores_kernel(
    const float* __restrict__ adjacency,  // [B, N, N]
    const float* __restrict__ support,    // [B, N, H]
    const float* __restrict__ b0,
    const float* __restrict__ g0,
    const float* __restrict__ be0,
    const float* __restrict__ b1,
    float* __restrict__ scores)           // [B, N]
{
    constexpr int APAD = 516;             // 512 + 4 pad
    constexpr int XPAD = 260;
    __shared__ float As[16 * APAD];       // 33 KB; reused as x[16][256] post-GEMM

    const int tid = threadIdx.x;
    const int b   = blockIdx.x >> 5;
    const int mt  = blockIdx.x & 31;

    // stage 16x512 adjacency tile
    const float4* src = (const float4*)(adjacency + ((long)b * Nn + mt * 16) * Nn);
    for (int i = tid; i < 16 * (Nn / 4); i += 256) {
        int r = i >> 7, c4 = i & 127;
        ((float4*)(As + r * APAD))[c4] = src[i];
    }
    __syncthreads();

    const int wave  = tid >> 5;
    const int lane  = tid & 31;
    const int row   = lane & 15;
    const int khalf = lane >> 4;
    const int colA  = wave * 32 + (lane & 15);
    const int colB  = colA + 16;
    const float* supb = support + (long)b * Nn * Hn;

    v8f acc0 = {}; v8f acc1 = {};
    for (int kb = 0; kb < Nn; kb += 64) {
        if (kb + 64 < Nn)   // pull the next streamed support block toward L2/L0
            __builtin_prefetch(supb + (long)(kb + 64 + 2 * khalf) * Hn + colA, 0, 1);
        #pragma unroll
        for (int k0 = kb; k0 < kb + 64; k0 += 4) {
            const int ka = k0 + 2 * khalf;
            v2f a;  a.x  = As[row * APAD + ka];
                    a.y  = As[row * APAD + ka + 1];
            const float* sp = supb + (long)ka * Hn;
            v2f bb0; bb0.x = sp[colA]; bb0.y = sp[Hn + colA];
            v2f bb1; bb1.x = sp[colB]; bb1.y = sp[Hn + colB];
            acc0 = WMMA_F32X4(a, bb0, acc0);
            acc1 = WMMA_F32X4(a, bb1, acc1);
        }
    }
    __syncthreads();                      // done with adjacency staging

    // transpose-store x (+b0) into LDS for row-wise LN
    float* xs = As;
    const int crow = (lane < 16) ? 0 : 8;
    const int ccol = lane & 15;
    #pragma unroll
    for (int r = 0; r < 8; ++r) {
        const int rr = crow + r;
        const int c0 = wave * 32 + ccol;
        xs[rr * XPAD + c0]      = acc0[r] + b0[c0];
        xs[rr * XPAD + c0 + 16] = acc1[r] + b0[c0 + 16];
    }
    __syncthreads();

    // LN + ReLU + b1 + row-sum; each wave handles rows {wave, wave+8}
    for (int rr = wave; rr < 16; rr += 8) {
        float s = 0.f, sq = 0.f, vals[8];
        #pragma unroll
        for (int i = 0; i < 8; ++i) {
            float v = xs[rr * XPAD + lane + 32 * i];
            vals[i] = v; s += v; sq += v * v;
        }
        #pragma unroll
        for (int off = 16; off >= 1; off >>= 1) {
            s  += __shfl_xor(s,  off, 32);
            sq += __shfl_xor(sq, off, 32);
        }
        const float mean = s * (1.f / Hn);
        const float var  = sq * (1.f / Hn) - mean * mean;
        const float rstd = rsqrtf(var + EPS);
        float sc = 0.f;
        #pragma unroll
        for (int i = 0; i < 8; ++i) {
            const int col = lane + 32 * i;
            float y = (vals[i] - mean) * rstd * g0[col] + be0[col];
            y = fmaxf(y, 0.f);
            sc += y + b1[col];
        }
        #pragma unroll
        for (int off = 16; off >= 1; off >>= 1) sc += __shfl_xor(sc, off, 32);
        if (lane == 0) scores[b * Nn + mt * 16 + rr] = sc;
    }
}

// ---------------------------------------------------------------------------
// Block-wide (sum, sum) reduction over 256 threads. rbuf must hold >= 18 f32.
// ---------------------------------------------------------------------------
__device__ __forceinline__ void block_reduce2(float a, float c, float* rbuf,
                                              int tid, float& oa, float& oc) {
    const int lane = tid & 31, wave = tid >> 5;
    #pragma unroll
    for (int off = 16; off >= 1; off >>= 1) {
        a += __shfl_xor(a, off, 32);
        c += __shfl_xor(c, off, 32);
    }
    if (lane == 0) { rbuf[wave] = a; rbuf[8 + wave] = c; }
    __syncthreads();
    if (wave == 0) {
        float aa = (lane < 8) ? rbuf[lane]     : 0.f;
        float cc = (lane < 8) ? rbuf[8 + lane] : 0.f;
        #pragma unroll
        for (int off = 4; off >= 1; off >>= 1) {
            aa += __shfl_xor(aa, off, 32);
            cc += __shfl_xor(cc, off, 32);
        }
        if (lane == 0) { rbuf[16] = aa; rbuf[17] = cc; }
    }
    __syncthreads();
    oa = rbuf[16]; oc = rbuf[17];
    __syncthreads();
}

// ---------------------------------------------------------------------------
// Kernel 3: per-batch argmax pooling + LN1 + W2 matvec + LN2 + head.
// grid = B blocks of 256 threads (thread == hidden channel).
// ---------------------------------------------------------------------------
__global__ __launch_bounds__(256) void pool_head_kernel(
    const float* __restrict__ adjacency,
    const float* __restrict__ support,
    const int*   __restrict__ stage_ids,
    const float* __restrict__ scores,
    const float* __restrict__ b0,  const float* __restrict__ g0, const float* __restrict__ be0,
    const float* __restrict__ b1,  const float* __restrict__ g1, const float* __restrict__ be1,
    const float* __restrict__ W2,  const float* __restrict__ b2,
    const float* __restrict__ g2,  const float* __restrict__ be2,
    const float* __restrict__ fcW, const float* __restrict__ fcb,
    float* __restrict__ outp)      // [B, 1]
{
    __shared__ float sc[Nn];
    __shared__ int   sid[Nn];
    __shared__ float arow[Nn];
    __shared__ float s2[Hn];
    __shared__ float rbuf[18];
    __shared__ float bredf[8];
    __shared__ int   bredi[8];
    __shared__ int   bestIdx;

    const int b   = blockIdx.x;
    const int tid = threadIdx.x;
    const int lane = tid & 31, wave = tid >> 5;

    for (int i = tid; i < Nn; i += 256) {
        sc[i]  = scores[b * Nn + i];
        sid[i] = stage_ids[b * Nn + i];
    }
    s2[tid] = 0.f;
    __syncthreads();

    const float* supb = support + (long)b * Nn * Hn;
    const float  NEGINF = -__builtin_inff();

    for (int s = 0; s < NSTAGE; ++s) {
        // ---- first-occurrence argmax over masked scores (jnp.argmax semantics)
        float best = NEGINF; int bi = 0x7fffffff;
        for (int i = tid; i < Nn; i += 256) {
            if (sid[i] == s) {
                float v = sc[i];
                if (v > best || (v == best && i < bi)) { best = v; bi = i; }
            }
        }
        #pragma unroll
        for (int off = 16; off >= 1; off >>= 1) {
            float ob = __shfl_xor(best, off, 32);
            int   oi = __shfl_xor(bi,   off, 32);
            if (ob > best || (ob == best && oi < bi)) { best = ob; bi = oi; }
        }
        if (lane == 0) { bredf[wave] = best; bredi[wave] = bi; }
        __syncthreads();
        if (tid == 0) {
            float bb = bredf[0]; int ii = bredi[0];
            for (int w = 1; w < 8; ++w)
                if (bredf[w] > bb || (bredf[w] == bb && bredi[w] < ii)) { bb = bredf[w]; ii = bredi[w]; }
            bestIdx = (bb == NEGINF) ? 0 : ii;   // empty group -> index 0
        }
        __syncthreads();
        const int rowIdx = bestIdx;

        // ---- recompute selected row: adj[b,rowIdx,:] @ support[b] + b0
        for (int i = tid; i < Nn; i += 256)
            arow[i] = adjacency[((long)b * Nn + rowIdx) * Nn + i];
        __syncthreads();

        float acc = b0[tid];
        for (int k = 0; k < Nn; ++k)
            acc = fmaf(arow[k], supb[(long)k * Hn + tid], acc);

        // ---- LN0 + ReLU + b1
        float mean, msq;
        block_reduce2(acc, acc * acc, rbuf, tid, mean, msq);
        mean *= (1.f / Hn); msq *= (1.f / Hn);
        float rstd = rsqrtf(msq - mean * mean + EPS);
        float y = fmaxf((acc - mean) * rstd * g0[tid] + be0[tid], 0.f);
        float outv = y + b1[tid];

        // ---- LN1 + ReLU on pooled row, accumulate stage sum
        block_reduce2(outv, outv * outv, rbuf, tid, mean, msq);
        mean *= (1.f / Hn); msq *= (1.f / Hn);
        rstd = rsqrtf(msq - mean * mean + EPS);
        float x1 = fmaxf((outv - mean) * rstd * g1[tid] + be1[tid], 0.f);
        s2[tid] += x1;
        __syncthreads();
    }

    // ---- row 0 of final_adj sums all stages: x0pre = (sum_s x1_s) @ W2 + b2
    float x0p = b2[tid];
    for (int k = 0; k < Hn; ++k)
        x0p = fmaf(s2[k], W2[k * Hn + tid], x0p);

    float mean, msq;
    block_reduce2(x0p, x0p * x0p, rbuf, tid, mean, msq);
    mean *= (1.f / Hn); msq *= (1.f / Hn);
    const float rstd = rsqrtf(msq - mean * mean + EPS);
    const float x0 = fmaxf((x0p - mean) * rstd * g2[tid] + be2[tid], 0.f);

    // ---- head: dot(x0, fcW) + fcb
    float part = x0 * fcW[tid], dummy;
    block_reduce2(part, 0.f, rbuf, tid, part, dummy);
    if (tid == 0) outp[b] = part + fcb[0];
}

// ---------------------------------------------------------------------------
extern "C" void kernel_launch(void* const* d_in, const int* in_sizes, int n_in,
                              void* d_out, int out_size, void* d_ws, size_t ws_size,
                              hipStream_t stream) {
    const float* adjacency = (const float*)d_in[0];
    const float* features  = (const float*)d_in[1];
    const int*   stage_ids = (const int*)  d_in[2];
    const float* W0  = (const float*)d_in[3];
    const float* b0  = (const float*)d_in[4];
    const float* b1  = (const float*)d_in[5];
    const float* W2  = (const float*)d_in[6];
    const float* b2  = (const float*)d_in[7];
    const float* g0  = (const float*)d_in[8];
    const float* be0 = (const float*)d_in[9];
    const float* g1  = (const float*)d_in[10];
    const float* be1 = (const float*)d_in[11];
    const float* g2  = (const float*)d_in[12];
    const float* be2 = (const float*)d_in[13];
    const float* fcW = (const float*)d_in[14];
    const float* fcb = (const float*)d_in[15];
    float* outp = (float*)d_out;

    // workspace: support [B*N*H] f32 (134 MB) + scores [B*N] f32 (0.5 MB)
    float* support = (float*)d_ws;
    float* scores  = support + (size_t)Bn * Nn * Hn;

    gemm1_kernel<<<(Bn * Nn) / 16, 256, 0, stream>>>(features, W0, support);
    gemm2_scores_kernel<<<Bn * (Nn / 16), 256, 0, stream>>>(
        adjacency, support, b0, g0, be0, b1, scores);
    pool_head_kernel<<<Bn, 256, 0, stream>>>(
        adjacency, support, stage_ids, scores,
        b0, g0, be0, b1, g1, be1, W2, b2, g2, be2, fcW, fcb, outp);
}